// MapLangSemanticFilter_10161892622960
// MI455X (gfx1250) — compile-verified
//
#include <hip/hip_runtime.h>

#define CIN   32
#define HW    65536
#define NEG   0.01f

typedef float v2f __attribute__((ext_vector_type(2)));
typedef float v8f __attribute__((ext_vector_type(8)));

// ---------------------------------------------------------------------------
// Kernel 1: cw = sigmoid(text @ W^T + b), L2-normalized per group row (1024)
// One block per group (G=8), 256 threads, each thread computes 4 outputs.
// ---------------------------------------------------------------------------
__global__ __launch_bounds__(256) void cw_kernel(const float* __restrict__ text,
                                                 const float* __restrict__ Wl,
                                                 const float* __restrict__ bl,
                                                 float* __restrict__ cw)
{
    __shared__ float t[256];
    __shared__ float red[9];
    const int g   = blockIdx.x;
    const int tid = threadIdx.x;
    t[tid] = text[g * 256 + tid];
    __syncthreads();

    float vals[4];
    float sq = 0.f;
    #pragma unroll
    for (int jj = 0; jj < 4; ++jj) {
        const int j = jj * 256 + tid;
        const float* __restrict__ w = Wl + j * 256;
        float s = bl[j];
        for (int k = 0; k < 256; ++k) s = fmaf(t[k], w[k], s);
        const float sig = 1.f / (1.f + expf(-s));
        vals[jj] = sig;
        sq += sig * sig;
    }
    // wave32 reduce, then cross-wave via LDS
    #pragma unroll
    for (int m = 16; m; m >>= 1) sq += __shfl_xor(sq, m);
    if ((tid & 31) == 0) red[tid >> 5] = sq;
    __syncthreads();
    if (tid == 0) {
        float tot = 0.f;
        for (int w = 0; w < 8; ++w) tot += red[w];
        red[8] = 1.f / fmaxf(sqrtf(tot), 1e-12f);
    }
    __syncthreads();
    const float rn = red[8];
    #pragma unroll
    for (int jj = 0; jj < 4; ++jj) cw[g * 1024 + jj * 256 + tid] = vals[jj] * rn;
}

// ---------------------------------------------------------------------------
// Conv passes. STORE=0: stats only (per-block partial sums / sumsq).
// STORE=1: recompute + leaky-relu + instance-norm + store.
// Block = 256 threads = 8 waves; each wave owns 256 spatial positions,
// processed in 4 staged chunks of 64. WMMA: D(16x16) = cw(16x4) x X(4x16).
// ---------------------------------------------------------------------------
template <int STORE>
__global__ __launch_bounds__(256) void conv_kernel(const float* __restrict__ in,
                                                   const float* __restrict__ cw,
                                                   const float* __restrict__ meanrstd,
                                                   float* __restrict__ partials,
                                                   float* __restrict__ out)
{
    __shared__ float lds[8 * 2048];          // 64 KB: 8 waves x (32ch x 64pos)
    const int tid  = threadIdx.x;
    const int wave = tid >> 5;
    const int lane = tid & 31;
    const int l16  = lane & 15;
    const int half = lane >> 4;              // 0: lanes 0-15, 1: lanes 16-31
    const int blk  = blockIdx.x;
    const int b    = blk >> 5;               // 32 chunks per image
    const int chunk = blk & 31;
    const int g    = b >> 2;                 // group = b / F, F = 4
    const int wbase = wave * 2048;
    const float* __restrict__ inB = in + (size_t)b * CIN * HW;

    // A fragments (per-wave constant): a0 -> o=0..15, a1 -> o=16..31
    v2f a0[8], a1[8];
    {
        const float* __restrict__ cwg = cw + g * 1024;
        const int r0 = l16, r1 = l16 + 16;
        #pragma unroll
        for (int k = 0; k < 8; ++k) {
            const int kb = 4 * k + 2 * half; // lanes>=16 hold K=2,3 of each step
            a0[k].x = cwg[r0 * 32 + kb];
            a0[k].y = cwg[r0 * 32 + kb + 1];
            a1[k].x = cwg[r1 * 32 + kb];
            a1[k].y = cwg[r1 * 32 + kb + 1];
        }
    }

    float mean0[8], rstd0[8], mean1[8], rstd1[8];
    float s0[8] = {0}, q0[8] = {0}, s1[8] = {0}, q1[8] = {0};
    if (STORE) {
        const float* __restrict__ mr = meanrstd + b * 64;   // [o]{mean,rstd}
        #pragma unroll
        for (int m = 0; m < 8; ++m) {
            const int o0 = m + 8 * half;                    // D row owned by lane
            mean0[m] = mr[o0 * 2];        rstd0[m] = mr[o0 * 2 + 1];
            mean1[m] = mr[(o0 + 16) * 2]; rstd1[m] = mr[(o0 + 16) * 2 + 1];
        }
    }

    const int wavePos = chunk * 2048 + wave * 256;
    for (int it = 0; it < 4; ++it) {
        const int pos0 = wavePos + it * 64;
        // Stage 32ch x 64pos via b128 loads: lanes 0-15 -> ch=2p, 16-31 -> 2p+1
        #pragma unroll
        for (int pr = 0; pr < 16; ++pr) {
            const int ch = 2 * pr + half;
            const float4 v = *reinterpret_cast<const float4*>(
                inB + (size_t)ch * HW + pos0 + 4 * l16);
            *reinterpret_cast<float4*>(&lds[wbase + ch * 64 + 4 * l16]) = v;
        }
        #pragma unroll
        for (int t = 0; t < 4; ++t) {
            const int col = t * 16 + l16;
            v8f acc0 = {}, acc1 = {};
            #pragma unroll
            for (int k = 0; k < 8; ++k) {
                const int kr = 4 * k + 2 * half;            // B: K rows per half
                v2f bb;
                bb.x = lds[wbase + kr * 64 + col];
                bb.y = lds[wbase + (kr + 1) * 64 + col];
                acc0 = __builtin_amdgcn_wmma_f32_16x16x4_f32(
                    false, a0[k], false, bb, (short)0, acc0, false, false);
                acc1 = __builtin_amdgcn_wmma_f32_16x16x4_f32(
                    false, a1[k], false, bb, (short)0, acc1, false, false);
            }
            #pragma unroll
            for (int m = 0; m < 8; ++m) {
                float y0 = acc0[m]; y0 = y0 > 0.f ? y0 : NEG * y0;
                float y1 = acc1[m]; y1 = y1 > 0.f ? y1 : NEG * y1;
                if (STORE) {
                    const int o0 = m + 8 * half;
                    // in-place: these columns were fully consumed by the k-loop
                    lds[wbase + o0 * 64 + col]        = (y0 - mean0[m]) * rstd0[m];
                    lds[wbase + (o0 + 16) * 64 + col] = (y1 - mean1[m]) * rstd1[m];
                } else {
                    s0[m] += y0; q0[m] += y0 * y0;
                    s1[m] += y1; q1[m] += y1 * y1;
                }
            }
        }
        if (STORE) {
            float* __restrict__ outB = out + (size_t)b * 32 * HW;
            #pragma unroll
            for (int pr = 0; pr < 16; ++pr) {
                const int ch = 2 * pr + half;
                const float4 v = *reinterpret_cast<const float4*>(
                    &lds[wbase + ch * 64 + 4 * l16]);
                *reinterpret_cast<float4*>(outB + (size_t)ch * HW + pos0 + 4 * l16) = v;
            }
        }
    }

    if (!STORE) {
        // reduce over the 16 columns each half-wave owns
        #pragma unroll
        for (int m = 0; m < 8; ++m) {
            #pragma unroll
            for (int off = 8; off; off >>= 1) {
                s0[m] += __shfl_xor(s0[m], off);
                q0[m] += __shfl_xor(q0[m], off);
                s1[m] += __shfl_xor(s1[m], off);
                q1[m] += __shfl_xor(q1[m], off);
            }
        }
        __syncthreads();                      // stage data dead; reuse LDS
        float* sumA = lds;                    // [wave][32]
        float* sqA  = lds + 256;              // [wave][32]
        if (l16 == 0) {
            #pragma unroll
            for (int m = 0; m < 8; ++m) {
                const int o0 = m + 8 * half;
                sumA[wave * 32 + o0]      = s0[m];
                sumA[wave * 32 + o0 + 16] = s1[m];
                sqA[wave * 32 + o0]       = q0[m];
                sqA[wave * 32 + o0 + 16]  = q1[m];
            }
        }
        __syncthreads();
        if (tid < 64) {
            const int o = tid & 31;
            float acc = 0.f;
            if (tid < 32) { for (int w = 0; w < 8; ++w) acc += sumA[w * 32 + o]; }
            else          { for (int w = 0; w < 8; ++w) acc += sqA[w * 32 + o]; }
            partials[blk * 64 + tid] = acc;   // [0:32]=sum, [32:64]=sumsq
        }
    }
}

// ---------------------------------------------------------------------------
// Kernel 3: fold 32 chunk-partials per (b,o) into mean / rstd.
// ---------------------------------------------------------------------------
__global__ __launch_bounds__(256) void finalize_kernel(const float* __restrict__ partials,
                                                       float* __restrict__ meanrstd)
{
    const int idx = blockIdx.x * 256 + threadIdx.x;   // b*32 + o, 1024 total
    if (idx >= 1024) return;
    const int b = idx >> 5, o = idx & 31;
    float S = 0.f, Q = 0.f;
    for (int c = 0; c < 32; ++c) {
        const int blk = b * 32 + c;
        S += partials[blk * 64 + o];
        Q += partials[blk * 64 + 32 + o];
    }
    const float mean = S * (1.f / 65536.f);
    const float var  = Q * (1.f / 65536.f) - mean * mean;
    meanrstd[idx * 2]     = mean;
    meanrstd[idx * 2 + 1] = rsqrtf(var + 1e-5f);
}

extern "C" void kernel_launch(void* const* d_in, const int* in_sizes, int n_in,
                              void* d_out, int out_size, void* d_ws, size_t ws_size,
                              hipStream_t stream)
{
    const float* input = (const float*)d_in[0];   // (32,32,256,256)
    const float* text  = (const float*)d_in[1];   // (8,256)
    const float* Wl    = (const float*)d_in[2];   // (1024,256)
    const float* bl    = (const float*)d_in[3];   // (1024,)
    float* out = (float*)d_out;

    float* cw       = (float*)d_ws;               // 8192 floats  (32 KB)
    float* partials = cw + 8192;                  // 1024*64 floats (256 KB)
    float* meanrstd = partials + 1024 * 64;       // 2048 floats  (8 KB)

    cw_kernel<<<8, 256, 0, stream>>>(text, Wl, bl, cw);
    conv_kernel<0><<<1024, 256, 0, stream>>>(input, cw, nullptr, partials, nullptr);
    finalize_kernel<<<4, 256, 0, stream>>>(partials, meanrstd);
    conv_kernel<1><<<1024, 256, 0, stream>>>(input, cw, meanrstd, partials, out);
}